// DiffusionNCA_fft2_68040871903467
// MI455X (gfx1250) — compile-verified
//
#include <hip/hip_runtime.h>
#include <math.h>

// ---------------------------------------------------------------------------
// DiffusionNCA fft2 step for MI455X (gfx1250, wave32, WMMA + async-to-LDS).
// Pipeline (steps == 1 per setup_inputs):
//   fft2 (DFT in LDS) -> pack dx (+alive ch63) -> reflect pad ->
//   dual 3x3 conv as one implicit WMMA f32 GEMM (K=576, N=128) ->
//   fc0 WMMA GEMM + bias + leaky_relu -> LayerNorm (block reduce + atomics) ->
//   fc1 WMMA GEMM -> stochastic masked complex update -> ifft2 -> out.
// Matrix math in fp32 via V_WMMA_F32_16X16X4_F32.  GEMM LDS staging uses
// GLOBAL_LOAD_ASYNC_TO_LDS_B128 double buffering (ASYNCcnt).  Both A and B
// are kept so each WMMA fragment is one contiguous ds_load_b64 (no repacking):
// A row-major [m][k], B N-major [n][k] (weights pre-transposed in k_prep),
// both with row stride 20 floats (16B-aligned b128 stores, conflict-free).
// ---------------------------------------------------------------------------

typedef __attribute__((ext_vector_type(2))) float v2f;
typedef __attribute__((ext_vector_type(8))) float v8f;

// Builtin signature (from hipcc diagnostic): pointee is a GCC vector of 4 ints,
// src in address_space(1) (global), dst in address_space(3) (LDS).
typedef int v4i __attribute__((vector_size(16)));
typedef __attribute__((address_space(1))) v4i* p1_v4i;
typedef __attribute__((address_space(3))) v4i* p3_v4i;

#define BATCH 32

#if defined(__has_builtin)
#if __has_builtin(__builtin_amdgcn_global_load_async_to_lds_b128)
#define HAS_ASYNC_LDS 1
#endif
#endif
#ifndef HAS_ASYNC_LDS
#define HAS_ASYNC_LDS 0
#endif

// Copy 16 contiguous bytes global -> LDS (per lane).
__device__ __forceinline__ void cp16(const float* __restrict__ g, float* l) {
#if HAS_ASYNC_LDS
  __builtin_amdgcn_global_load_async_to_lds_b128(
      (p1_v4i)(void*)g, (p3_v4i)(void*)l, /*imm_offset=*/0, /*cpol=*/0);
#else
  *(float4*)l = *(const float4*)g;
#endif
}

__device__ __forceinline__ void wait_async() {
#if HAS_ASYNC_LDS
  asm volatile("s_wait_asynccnt 0x0" ::: "memory");
#endif
}

__device__ __forceinline__ v8f wmma4(v2f a, v2f b, v8f c) {
  // D(16x16,f32) = A(16x4,f32) * B(4x16,f32) + C
  return __builtin_amdgcn_wmma_f32_16x16x4_f32(
      /*neg_a=*/false, a, /*neg_b=*/false, b,
      /*c_mod=*/(short)0, c, /*reuse_a=*/false, /*reuse_b=*/false);
}

// ---------------------------------------------------------------------------
// Prep: DFT tables (symmetric 64x64 cos/sin), transposed conv weights
// WcatT[oc][tap*64+ic] (oc 0..63 = conv0, 64..127 = conv1), Bcat[128],
// and transposed fc weights FC0T[256][192], FC1T[64][256].
// ---------------------------------------------------------------------------
__global__ void k_prep(float* __restrict__ Fr, float* __restrict__ Fi,
                       float* __restrict__ WcatT, float* __restrict__ Bcat,
                       float* __restrict__ FC0T, float* __restrict__ FC1T,
                       const float* __restrict__ p0w, const float* __restrict__ p0b,
                       const float* __restrict__ p1w, const float* __restrict__ p1b,
                       const float* __restrict__ fc0w, const float* __restrict__ fc1w) {
  int idx = blockIdx.x * blockDim.x + threadIdx.x;
  int stride = gridDim.x * blockDim.x;
  for (int e = idx; e < 64 * 64; e += stride) {
    int k = e >> 6, n = e & 63;
    int m = (k * n) & 63;                       // exact phase mod N
    float ang = -3.14159265358979323846f * (float)m / 32.0f;
    Fr[e] = cosf(ang);
    Fi[e] = sinf(ang);
  }
  for (int e = idx; e < 128 * 576; e += stride) {
    int oc = e / 576, kk = e - oc * 576;
    int tap = kk >> 6, ic = kk & 63;
    int ky = tap / 3, kx = tap - ky * 3;
    const float* w = (oc < 64) ? p0w : p1w;
    int o = oc & 63;
    WcatT[e] = w[((o * 64 + ic) * 3 + ky) * 3 + kx];
  }
  for (int e = idx; e < 128; e += stride)
    Bcat[e] = (e < 64) ? p0b[e] : p1b[e - 64];
  for (int e = idx; e < 256 * 192; e += stride) {   // FC0T[n][k] = fc0w[k][n]
    int n = e / 192, k = e - n * 192;
    FC0T[e] = fc0w[(size_t)k * 256 + n];
  }
  for (int e = idx; e < 64 * 256; e += stride) {    // FC1T[n][k] = fc1w[k][n]
    int n = e >> 8, k = e & 255;
    FC1T[e] = fc1w[(size_t)k * 64 + n];
  }
}

// ---------------------------------------------------------------------------
// Forward fft2: one block per (b,c) plane.  X[b,c,i,j] = x[b,j,i,c] (real).
// XF[b,c,k1,k2] stored channel-last at XFr/XFi[((b*64+k1)*64+k2)*32+c].
// DFT matrix is symmetric -> F tables always read coalesced/broadcast.
// ---------------------------------------------------------------------------
__global__ __launch_bounds__(256) void k_fft_fwd(
    const float* __restrict__ x, const float* __restrict__ gFr,
    const float* __restrict__ gFi, float* __restrict__ XFr,
    float* __restrict__ XFi) {
  __shared__ float Pr[64][65];                  // padded: lane-varying row store
  __shared__ float Trs[64][64], Tis[64][64];
  const int tid = threadIdx.x;
  const int plane = blockIdx.x;
  const int b = plane >> 5, c = plane & 31;
  for (int e = tid; e < 4096; e += 256) {
    int i = e & 63, j = e >> 6;
    Pr[i][j] = x[((size_t)(b * 64 + j) * 64 + i) * 32 + c];
  }
  __syncthreads();
  // stage 1: transform over j
  for (int e = tid; e < 4096; e += 256) {
    int i = e >> 6, k2 = e & 63;                // i warp-uniform, k2 = lane
    float sr = 0.f, si = 0.f;
    for (int j = 0; j < 64; ++j) {
      float p = Pr[i][j];                       // LDS broadcast
      float fr = gFr[j * 64 + k2];              // coalesced (F symmetric)
      float fi = gFi[j * 64 + k2];
      sr += p * fr; si += p * fi;
    }
    Trs[i][k2] = sr; Tis[i][k2] = si;
  }
  __syncthreads();
  // stage 2: transform over i
  for (int e = tid; e < 4096; e += 256) {
    int k1 = e >> 6, k2 = e & 63;
    float ar = 0.f, ai = 0.f;
    for (int i = 0; i < 64; ++i) {
      float fr = gFr[i * 64 + k1];              // scalar broadcast
      float fi = gFi[i * 64 + k1];
      float tr = Trs[i][k2], ti = Tis[i][k2];
      ar += fr * tr - fi * ti;
      ai += fr * ti + fi * tr;
    }
    size_t o = ((size_t)(b * 64 + k1) * 64 + k2) * 32 + c;
    XFr[o] = ar; XFi[o] = ai;
  }
}

// ---------------------------------------------------------------------------
// Pack dx: DX[b][i][j][ch] = {Re(c), Im(c-32)}; ch 63 overwritten with alive.
// Also copies into Y[...,0:64] (fc0 A-operand, ld=192).
// ---------------------------------------------------------------------------
__global__ void k_pack_dx(const float* __restrict__ XFr,
                          const float* __restrict__ XFi,
                          float* __restrict__ DX, float* __restrict__ Y) {
  int idx = blockIdx.x * 256 + threadIdx.x;
  if (idx >= BATCH * 64 * 64 * 64) return;
  int ch = idx & 63, j = (idx >> 6) & 63, i = (idx >> 12) & 63, b = idx >> 18;
  size_t plane = ((size_t)(b * 64 + i) * 64 + j) * 32;
  float v = (ch < 32) ? XFr[plane + ch] : XFi[plane + (ch - 32)];
  if (ch == 63) v = (2.0f - (float)(i + j) * (1.0f / 63.0f)) * 0.5f;  // alive
  DX[idx] = v;
  Y[((size_t)b * 4096 + (i * 64 + j)) * 192 + ch] = v;
}

// Reflect pad 1 on both spatial dims: DXP[b][66][66][64].
__global__ void k_pad(const float* __restrict__ DX, float* __restrict__ DXP) {
  int idx = blockIdx.x * 256 + threadIdx.x;
  if (idx >= BATCH * 66 * 66 * 64) return;
  int ch = idx & 63;
  int rest = idx >> 6;
  int pj = rest % 66; rest /= 66;
  int pi = rest % 66; int b = rest / 66;
  int qi = (pi == 0) ? 1 : ((pi == 65) ? 62 : pi - 1);
  int qj = (pj == 0) ? 1 : ((pj == 65) ? 62 : pj - 1);
  DXP[idx] = DX[(((size_t)b * 64 + qi) * 64 + qj) * 64 + ch];
}

// ---------------------------------------------------------------------------
// Dual conv as implicit GEMM: M=4096 spatial, N=128 (conv0|conv1), K=576.
// A gathered from reflect-padded channel-last input (per-lane async copies),
// B = pre-transposed weights staged N-major.  Double-buffered async staging.
// Block = 64x128 C-tile, 8 waves x 4 accumulators.
// ---------------------------------------------------------------------------
__global__ __launch_bounds__(256) void k_conv(
    const float* __restrict__ dxp, const float* __restrict__ WcatT,
    const float* __restrict__ Bcat, float* __restrict__ Y) {
  __shared__ __align__(16) float Asm[2][64][20];    // [m][k], stride 20
  __shared__ __align__(16) float Bst[2][128][20];   // [n][k], stride 20
  const int tid = threadIdx.x;
  const int lane = tid & 31, wid = tid >> 5;
  const int tm = wid & 3, tn0 = (wid >> 2) * 4;
  const int m0 = blockIdx.x * 64;
  const int b = blockIdx.y;
  const float* dp = dxp + (size_t)b * 66 * 66 * 64;
  v8f zero = {0.f, 0.f, 0.f, 0.f, 0.f, 0.f, 0.f, 0.f};
  v8f acc[4]; acc[0] = zero; acc[1] = zero; acc[2] = zero; acc[3] = zero;
  const int arow = tid >> 2, acb = (tid & 3) * 4;
  const int ai = (m0 + arow) >> 6, aj = (m0 + arow) & 63;
  const int bn = tid >> 1, bcb = (tid & 1) * 8;      // B: 128 rows x 16 k
  const int kb = (lane >> 4) * 2, rl = lane & 15;
  const float* Bg = WcatT + (size_t)bn * 576 + bcb;

  auto prefetch = [&](int kq, int buf) {
    int tap = kq >> 6, icb = kq & 63;
    int ky = tap / 3, kx = tap - ky * 3;
    cp16(dp + ((size_t)(ai + ky) * 66 + (aj + kx)) * 64 + icb + acb,
         &Asm[buf][arow][acb]);
    cp16(Bg + kq,     &Bst[buf][bn][bcb]);
    cp16(Bg + kq + 4, &Bst[buf][bn][bcb + 4]);
  };

  prefetch(0, 0);
  for (int k0 = 0; k0 < 576; k0 += 16) {
    const int cur = (k0 >> 4) & 1;
    wait_async();           // my async writes landed
    __syncthreads();        // everyone's writes visible; prev compute done
    if (k0 + 16 < 576) prefetch(k0 + 16, cur ^ 1);
#pragma unroll
    for (int kk = 0; kk < 16; kk += 4) {
      v2f a = *(const v2f*)&Asm[cur][tm * 16 + rl][kk + kb];
#pragma unroll
      for (int t = 0; t < 4; ++t) {
        v2f bb = *(const v2f*)&Bst[cur][(tn0 + t) * 16 + rl][kk + kb];
        acc[t] = wmma4(a, bb, acc[t]);
      }
    }
  }
  const int rowoff = (lane & 16) ? 8 : 0, cl = lane & 15;
#pragma unroll
  for (int t = 0; t < 4; ++t) {
#pragma unroll
    for (int r8 = 0; r8 < 8; ++r8) {
      int row = m0 + tm * 16 + r8 + rowoff;
      int col = (tn0 + t) * 16 + cl;
      Y[((size_t)b * 4096 + row) * 192 + 64 + col] = acc[t][r8] + Bcat[col];
    }
  }
}

// ---------------------------------------------------------------------------
// Generic batched fp32 WMMA GEMM with pre-transposed B:
//   C[z](MxN) (=) A[z](MxK, row-major) * BT(NxK, row-major)^T  [+bias][act].
// Double-buffered async LDS staging; block = 64x64 tile, 8 waves x 2 accums.
// K, lda, ldbt multiples of 16; M,N multiples of 64.
// ---------------------------------------------------------------------------
__global__ __launch_bounds__(256) void k_gemm(
    const float* __restrict__ A, int lda, size_t sA,
    const float* __restrict__ BT, int ldbt,
    float* __restrict__ C, int ldc, size_t sC,
    int K, const float* __restrict__ bias, int act) {
  __shared__ __align__(16) float Asm[2][64][20];    // [m][k]
  __shared__ __align__(16) float Bst[2][64][20];    // [n][k]
  const int tid = threadIdx.x;
  const int lane = tid & 31, wid = tid >> 5;
  const int tm = wid & 3, tn0 = (wid >> 2) * 2;
  const int m0 = blockIdx.x * 64, n0 = blockIdx.y * 64;
  A += (size_t)blockIdx.z * sA;
  C += (size_t)blockIdx.z * sC;
  v8f zero = {0.f, 0.f, 0.f, 0.f, 0.f, 0.f, 0.f, 0.f};
  v8f acc0 = zero, acc1 = zero;
  const int arow = tid >> 2, acb = (tid & 3) * 4;   // A: 64 rows x 16 k
  const int kb = (lane >> 4) * 2, rl = lane & 15;
  const float* Ag = A + (size_t)(m0 + arow) * lda + acb;
  const float* Bg = BT + (size_t)(n0 + arow) * ldbt + acb;  // B: 64 rows x 16 k

  cp16(Ag, &Asm[0][arow][acb]);
  cp16(Bg, &Bst[0][arow][acb]);
  for (int k0 = 0; k0 < K; k0 += 16) {
    const int cur = (k0 >> 4) & 1;
    wait_async();
    __syncthreads();
    if (k0 + 16 < K) {
      cp16(Ag + k0 + 16, &Asm[cur ^ 1][arow][acb]);
      cp16(Bg + k0 + 16, &Bst[cur ^ 1][arow][acb]);
    }
#pragma unroll
    for (int kk = 0; kk < 16; kk += 4) {
      v2f a  = *(const v2f*)&Asm[cur][tm * 16 + rl][kk + kb];
      v2f b0 = *(const v2f*)&Bst[cur][tn0 * 16 + rl][kk + kb];
      v2f b1 = *(const v2f*)&Bst[cur][(tn0 + 1) * 16 + rl][kk + kb];
      acc0 = wmma4(a, b0, acc0);
      acc1 = wmma4(a, b1, acc1);
    }
  }
  const int rowoff = (lane & 16) ? 8 : 0, cl = lane & 15;
#pragma unroll
  for (int r8 = 0; r8 < 8; ++r8) {
    int row = m0 + tm * 16 + r8 + rowoff;
    int c0 = n0 + tn0 * 16 + cl;
    int c1 = c0 + 16;
    float v0 = acc0[r8], v1 = acc1[r8];
    if (bias) { v0 += bias[c0]; v1 += bias[c1]; }
    if (act)  { v0 = v0 > 0.f ? v0 : 0.01f * v0; v1 = v1 > 0.f ? v1 : 0.01f * v1; }
    C[(size_t)row * ldc + c0] = v0;
    C[(size_t)row * ldc + c1] = v1;
  }
}

// ---------------------------------------------------------------------------
// LayerNorm over 1M elements per batch: zero -> block reduce + atomics -> apply.
// H layout [b][i][j][k]; ln weights read transposed: ln_w[j][i][k].
// ---------------------------------------------------------------------------
__global__ void k_ln_zero(float* stats) { stats[threadIdx.x] = 0.0f; }

__global__ __launch_bounds__(256) void k_ln_reduce(const float* __restrict__ H,
                                                   float* __restrict__ stats) {
  int b = blockIdx.x, chunk = blockIdx.y, tid = threadIdx.x;
  size_t base = ((size_t)b << 20) + ((size_t)chunk << 16) + tid;
  float s = 0.f, s2 = 0.f;
  for (int t = 0; t < 256; ++t) {
    float v = H[base + (size_t)t * 256];
    s += v; s2 += v * v;
  }
  for (int off = 16; off > 0; off >>= 1) {
    s  += __shfl_down(s,  off, 32);
    s2 += __shfl_down(s2, off, 32);
  }
  __shared__ float ps[8], ps2[8];
  int lane = tid & 31, wid = tid >> 5;
  if (lane == 0) { ps[wid] = s; ps2[wid] = s2; }
  __syncthreads();
  if (tid == 0) {
    float a = 0.f, a2 = 0.f;
    for (int w = 0; w < 8; ++w) { a += ps[w]; a2 += ps2[w]; }
    atomicAdd(&stats[b * 2], a);
    atomicAdd(&stats[b * 2 + 1], a2);
  }
}

__global__ void k_ln_apply(float* __restrict__ H, const float* __restrict__ stats,
                           const float* __restrict__ lnw, const float* __restrict__ lnb) {
  size_t idx = (size_t)blockIdx.x * 256 + threadIdx.x;
  int b = (int)(idx >> 20);
  int rem = (int)(idx & 1048575);
  int k = rem & 255, jj = (rem >> 8) & 63, ii = rem >> 14;
  const float inv = 1.0f / 1048576.0f;
  float mu  = stats[b * 2] * inv;
  float var = stats[b * 2 + 1] * inv - mu * mu;
  float rs = rsqrtf(var + 1e-5f);
  size_t wi = ((size_t)(jj * 64 + ii)) * 256 + k;   // transposed spatial access
  H[idx] = (H[idx] - mu) * rs * lnw[wi] + lnb[wi];
}

// ---------------------------------------------------------------------------
// Stochastic masked complex update: XF += mask * D (mask spatially transposed).
// ---------------------------------------------------------------------------
__global__ void k_update(const float* __restrict__ Dbuf,
                         const float* __restrict__ su,
                         float* __restrict__ XFr, float* __restrict__ XFi) {
  int idx = blockIdx.x * 256 + threadIdx.x;
  if (idx >= BATCH * 64 * 64 * 64) return;
  int c2 = idx & 63, j = (idx >> 6) & 63, i = (idx >> 12) & 63, b = idx >> 18;
  float u = su[((size_t)b * 64 + j) * 64 + i];
  float m = (u > 0.5f) ? 1.0f : 0.0f;
  float d = Dbuf[idx] * m;
  size_t o = ((size_t)(b * 64 + i) * 64 + j) * 32 + (c2 & 31);
  if (c2 < 32) XFr[o] += d;
  else         XFi[o] += d;
}

// ---------------------------------------------------------------------------
// Inverse fft2 (+1/4096 scale); writes interleaved complex64 output
// out[b][h=j][w=i][c] = {re, im}.
// ---------------------------------------------------------------------------
__global__ __launch_bounds__(256) void k_fft_inv(
    const float* __restrict__ XFr, const float* __restrict__ XFi,
    const float* __restrict__ gFr, const float* __restrict__ gFi,
    float* __restrict__ out) {
  __shared__ float Pr[64][64], Pi[64][64];
  __shared__ float Ur[64][64], Ui[64][64];
  const int tid = threadIdx.x;
  const int plane = blockIdx.x;
  const int b = plane >> 5, c = plane & 31;
  for (int e = tid; e < 4096; e += 256) {
    int k1 = e >> 6, k2 = e & 63;
    size_t o = ((size_t)(b * 64 + k1) * 64 + k2) * 32 + c;
    Pr[k1][k2] = XFr[o];
    Pi[k1][k2] = XFi[o];
  }
  __syncthreads();
  // stage 1: inverse transform over k2 (uses conj(F) = Fr - i*Fi)
  for (int e = tid; e < 4096; e += 256) {
    int k1 = e >> 6, j = e & 63;
    float ur = 0.f, ui = 0.f;
    for (int k2 = 0; k2 < 64; ++k2) {
      float pr = Pr[k1][k2], pi = Pi[k1][k2];       // LDS broadcast
      float fr = gFr[k2 * 64 + j], fi = gFi[k2 * 64 + j];  // coalesced
      ur += pr * fr + pi * fi;
      ui += pi * fr - pr * fi;
    }
    Ur[k1][j] = ur; Ui[k1][j] = ui;
  }
  __syncthreads();
  // stage 2: inverse transform over k1
  const float scale = 1.0f / 4096.0f;
  for (int e = tid; e < 4096; e += 256) {
    int j = e >> 6, i = e & 63;
    float yr = 0.f, yi = 0.f;
    for (int k1 = 0; k1 < 64; ++k1) {
      float ur = Ur[k1][j], ui = Ui[k1][j];         // LDS broadcast
      float fr = gFr[k1 * 64 + i], fi = gFi[k1 * 64 + i];  // coalesced
      yr += ur * fr + ui * fi;
      yi += ui * fr - ur * fi;
    }
    size_t o = ((size_t)(b * 64 + j) * 64 + i) * 32 + c;
    out[2 * o]     = yr * scale;
    out[2 * o + 1] = yi * scale;
  }
}

// ---------------------------------------------------------------------------
extern "C" void kernel_launch(void* const* d_in, const int* in_sizes, int n_in,
                              void* d_out, int out_size, void* d_ws, size_t ws_size,
                              hipStream_t stream) {
  (void)in_sizes; (void)n_in; (void)out_size; (void)ws_size;
  const float* x    = (const float*)d_in[0];
  const float* p0w  = (const float*)d_in[1];
  const float* p0b  = (const float*)d_in[2];
  const float* p1w  = (const float*)d_in[3];
  const float* p1b  = (const float*)d_in[4];
  const float* fc0w = (const float*)d_in[5];
  const float* fc0b = (const float*)d_in[6];
  const float* fc1w = (const float*)d_in[7];
  const float* lnw  = (const float*)d_in[8];
  const float* lnb  = (const float*)d_in[9];
  const float* su   = (const float*)d_in[10];
  float* out = (float*)d_out;

  float* ws = (float*)d_ws;
  size_t off = 0;
  float* FR    = ws + off; off += 4096;
  float* FI    = ws + off; off += 4096;
  float* WCATT = ws + off; off += 128 * 576;
  float* BCAT  = ws + off; off += 128;
  float* FC0T  = ws + off; off += 256 * 192;
  float* FC1T  = ws + off; off += 64 * 256;
  float* STAT  = ws + off; off += 64;
  float* XFR   = ws + off; off += (size_t)BATCH * 64 * 64 * 32;
  float* XFI   = ws + off; off += (size_t)BATCH * 64 * 64 * 32;
  float* DX    = ws + off; off += (size_t)BATCH * 64 * 64 * 64;
  float* DXP   = ws + off; off += (size_t)BATCH * 66 * 66 * 64;
  float* Ybuf  = ws + off; off += (size_t)BATCH * 4096 * 192;
  float* Hbuf  = ws + off; off += (size_t)BATCH * 4096 * 256;
  float* Dbuf  = ws + off; off += (size_t)BATCH * 4096 * 64;

  k_prep<<<64, 256, 0, stream>>>(FR, FI, WCATT, BCAT, FC0T, FC1T,
                                 p0w, p0b, p1w, p1b, fc0w, fc1w);
  k_fft_fwd<<<1024, 256, 0, stream>>>(x, FR, FI, XFR, XFI);

  // ---- one NCA step (steps == 1 in setup_inputs) ----
  k_pack_dx<<<32768, 256, 0, stream>>>(XFR, XFI, DX, Ybuf);
  k_pad<<<34848, 256, 0, stream>>>(DX, DXP);
  k_conv<<<dim3(64, 32), 256, 0, stream>>>(DXP, WCATT, BCAT, Ybuf);
  k_gemm<<<dim3(64, 4, 32), 256, 0, stream>>>(
      Ybuf, 192, (size_t)4096 * 192, FC0T, 192,
      Hbuf, 256, (size_t)4096 * 256, 192, fc0b, 1);
  k_ln_zero<<<1, 64, 0, stream>>>(STAT);
  k_ln_reduce<<<dim3(32, 16), 256, 0, stream>>>(Hbuf, STAT);
  k_ln_apply<<<131072, 256, 0, stream>>>(Hbuf, STAT, lnw, lnb);
  k_gemm<<<dim3(64, 1, 32), 256, 0, stream>>>(
      Hbuf, 256, (size_t)4096 * 256, FC1T, 256,
      Dbuf, 64, (size_t)4096 * 64, 256, nullptr, 0);
  k_update<<<32768, 256, 0, stream>>>(Dbuf, su, XFR, XFI);
  // ---- end step ----

  k_fft_inv<<<1024, 256, 0, stream>>>(XFR, XFI, FR, FI, out);
}